// Net_37271726195388
// MI455X (gfx1250) — compile-verified
//
#include <hip/hip_runtime.h>
#include <hip/hip_bf16.h>
#include <math.h>

#define N_NODES 100000
#define N_EDGES 800000
#define ANNOT   92
#define HID     64
#define N_STEPS 8
#define N_IDX   20000

#define NTILES   (N_NODES / 16)      // 6250 row tiles of 16 (exact)
#define WPB      4                   // waves per block
#define TPB      (WPB * 32)          // 128 threads

// Transposed LDS weight layouts [n][k]; pitches chosen for conflict-free
// b64 reads: pitch mod 64 = {28, 4} -> 16 lanes hit 16 distinct banks,
// upper half-wave (+2 dwords) hits the disjoint odd-pair set.
#define RED_PITCH 92                 // reduce: k in [0,92)
#define MST_PITCH 68                 // mstep:  k in [0,64) + pad

typedef __attribute__((ext_vector_type(2))) float v2f;
typedef __attribute__((ext_vector_type(8))) float v8f;

__device__ __forceinline__ v8f vzero8() {
    v8f z = {0.f, 0.f, 0.f, 0.f, 0.f, 0.f, 0.f, 0.f};
    return z;
}

// D = A(16x4,f32) * B(4x16,f32) + C(16x16,f32)
__device__ __forceinline__ v8f wmma4(v2f a, v2f b, v8f c) {
    return __builtin_amdgcn_wmma_f32_16x16x4_f32(
        /*neg_a=*/false, a, /*neg_b=*/false, b,
        /*c_mod=*/(short)0, c, /*reuse_a=*/false, /*reuse_b=*/false);
}

__device__ __forceinline__ float sigmoidf_(float x) {
    return 1.0f / (1.0f + expf(-x));
}

// Async-copy one f32 from global to LDS (CDNA5 path, tracked by ASYNCcnt).
__device__ __forceinline__ void async_g2l_b32(const float* gp, const float* lp) {
    const unsigned ldsoff = (unsigned)(size_t)lp;   // low 32 bits of LDS aperture addr = LDS offset
    asm volatile("global_load_async_to_lds_b32 %0, %1, off"
                 :: "v"(ldsoff), "v"(gp)
                 : "memory");
}

__device__ __forceinline__ void async_wait_and_barrier() {
    asm volatile("s_wait_asynccnt 0x0" ::: "memory");
    __syncthreads();
}

// -------------------------------------------------------------------
// h[N,64] = x[N,92] @ W_red[92,64] + b_red
// One wave32 per 16-row tile; W_red staged TRANSPOSED in LDS (async copy)
// so a B-fragment is one contiguous ds_load_b64.
// -------------------------------------------------------------------
__global__ __launch_bounds__(TPB)
void reduce_kernel(const float* __restrict__ x, const float* __restrict__ Wred,
                   const float* __restrict__ bred, float* __restrict__ h) {
    __shared__ float shW[HID * RED_PITCH];   // [n][k]

    // Stage W_red[k][n] -> shW[n][k], uniform across block.
    for (int i = threadIdx.x; i < ANNOT * HID; i += TPB) {
        const int k = i >> 6;
        const int n = i & 63;
        async_g2l_b32(Wred + i, &shW[n * RED_PITCH + k]);
    }
    async_wait_and_barrier();

    const int tile = blockIdx.x * WPB + (threadIdx.x >> 5);
    if (tile < NTILES) {
        const int lane = threadIdx.x & 31;
        const int m16  = lane & 15;      // row within tile / col within ntile
        const int g    = lane >> 4;      // K-half selector
        const int row0 = tile * 16;

        v8f acc[4];
        #pragma unroll
        for (int n = 0; n < 4; ++n) acc[n] = vzero8();

        #pragma unroll 4
        for (int k = 0; k < ANNOT; k += 4) {
            const v2f a = *(const v2f*)(x + (size_t)(row0 + m16) * ANNOT + k + 2 * g);
            #pragma unroll
            for (int n = 0; n < 4; ++n) {
                const int ncol = n * 16 + m16;
                const v2f b = *(const v2f*)(&shW[ncol * RED_PITCH + k + 2 * g]);
                acc[n] = wmma4(a, b, acc[n]);
            }
        }

        #pragma unroll
        for (int n = 0; n < 4; ++n) {
            const int ncol = n * 16 + m16;
            const float bias = bred[ncol];
            #pragma unroll
            for (int r = 0; r < 8; ++r) {
                const int M = r + 8 * g;
                h[(size_t)(row0 + M) * HID + ncol] = acc[n][r] + bias;
            }
        }
    }
}

// -------------------------------------------------------------------
// m[N,64] = h[N,64] @ Wt[64,64] ; Wt staged TRANSPOSED in LDS (async copy).
// -------------------------------------------------------------------
__global__ __launch_bounds__(TPB)
void mstep_kernel(const float* __restrict__ h, const float* __restrict__ Wt,
                  float* __restrict__ m) {
    __shared__ float shW[HID * MST_PITCH];   // [n][k]

    for (int i = threadIdx.x; i < HID * HID; i += TPB) {
        const int k = i >> 6;
        const int n = i & 63;
        async_g2l_b32(Wt + i, &shW[n * MST_PITCH + k]);
    }
    async_wait_and_barrier();

    const int tile = blockIdx.x * WPB + (threadIdx.x >> 5);
    if (tile < NTILES) {
        const int lane = threadIdx.x & 31;
        const int m16  = lane & 15;
        const int g    = lane >> 4;
        const int row0 = tile * 16;

        v8f acc[4];
        #pragma unroll
        for (int n = 0; n < 4; ++n) acc[n] = vzero8();

        #pragma unroll 4
        for (int k = 0; k < HID; k += 4) {
            const v2f a = *(const v2f*)(h + (size_t)(row0 + m16) * HID + k + 2 * g);
            #pragma unroll
            for (int n = 0; n < 4; ++n) {
                const int ncol = n * 16 + m16;
                const v2f b = *(const v2f*)(&shW[ncol * MST_PITCH + k + 2 * g]);
                acc[n] = wmma4(a, b, acc[n]);
            }
        }

        #pragma unroll
        for (int n = 0; n < 4; ++n) {
            const int ncol = n * 16 + m16;
            #pragma unroll
            for (int r = 0; r < 8; ++r) {
                const int M = r + 8 * g;
                m[(size_t)(row0 + M) * HID + ncol] = acc[n][r];
            }
        }
    }
}

// -------------------------------------------------------------------
// aggr = 0
// -------------------------------------------------------------------
__global__ void zero_kernel(float4* __restrict__ p, int n4) {
    const int i = blockIdx.x * blockDim.x + threadIdx.x;
    if (i < n4) p[i] = make_float4(0.f, 0.f, 0.f, 0.f);
}

// -------------------------------------------------------------------
// aggr[dst[e]] += m[src[e]]   (16 threads per edge, float4 each)
// -------------------------------------------------------------------
__global__ void scatter_kernel(const float* __restrict__ m, const int* __restrict__ src,
                               const int* __restrict__ dst, float* __restrict__ aggr) {
    const long long t = (long long)blockIdx.x * blockDim.x + threadIdx.x;
    if (t >= (long long)N_EDGES * 16) return;
    const int e = (int)(t >> 4);
    const int c = (int)(t & 15) * 4;
    const int s = src[e];
    const int d = dst[e];
    const float4 v = *(const float4*)(m + (size_t)s * HID + c);
    float* p = aggr + (size_t)d * HID + c;
    atomicAdd(p + 0, v.x);
    atomicAdd(p + 1, v.y);
    atomicAdd(p + 2, v.z);
    atomicAdd(p + 3, v.w);
}

// -------------------------------------------------------------------
// GRU: gi = aggr @ w_ih.T + b_ih ; gh = h @ w_hh.T + b_hh
// r = sig(i_r+h_r); z = sig(i_z+h_z); n = tanh(i_n + r*h_n)
// h = (1-z)*n + z*h   (in place; wave owns its 16 rows)
// gg: 0=r, 1=z, 2=n (gate column block in [0,192))
// -------------------------------------------------------------------
__device__ __forceinline__ void gate_gemm(const float* __restrict__ aggr,
                                          const float* __restrict__ h,
                                          const float* __restrict__ w_ih,
                                          const float* __restrict__ w_hh,
                                          int row0, int m16, int g, int gg,
                                          v8f acc_i[4], v8f acc_h[4]) {
    #pragma unroll
    for (int n = 0; n < 4; ++n) { acc_i[n] = vzero8(); acc_h[n] = vzero8(); }
    #pragma unroll 4
    for (int k = 0; k < HID; k += 4) {
        const v2f aa = *(const v2f*)(aggr + (size_t)(row0 + m16) * HID + k + 2 * g);
        const v2f ah = *(const v2f*)(h    + (size_t)(row0 + m16) * HID + k + 2 * g);
        #pragma unroll
        for (int n = 0; n < 4; ++n) {
            const int c = gg * 64 + n * 16 + m16;
            // B[k][c] = w[c][k]  -> contiguous float2 along k
            const v2f bi = *(const v2f*)(w_ih + (size_t)c * HID + k + 2 * g);
            const v2f bh = *(const v2f*)(w_hh + (size_t)c * HID + k + 2 * g);
            acc_i[n] = wmma4(aa, bi, acc_i[n]);
            acc_h[n] = wmma4(ah, bh, acc_h[n]);
        }
    }
}

__global__ __launch_bounds__(TPB)
void gru_kernel(const float* __restrict__ aggr, float* __restrict__ h,
                const float* __restrict__ w_ih, const float* __restrict__ w_hh,
                const float* __restrict__ b_ih, const float* __restrict__ b_hh) {
    const int tile = blockIdx.x * WPB + (threadIdx.x >> 5);
    if (tile >= NTILES) return;
    const int lane = threadIdx.x & 31;
    const int m16  = lane & 15;
    const int g    = lane >> 4;
    const int row0 = tile * 16;

    v8f ai[4], ah[4];

    // r gate
    gate_gemm(aggr, h, w_ih, w_hh, row0, m16, g, /*gg=*/0, ai, ah);
    v8f rg[4];
    #pragma unroll
    for (int n = 0; n < 4; ++n) {
        const int c = 0 * 64 + n * 16 + m16;
        const float bi = b_ih[c], bh = b_hh[c];
        #pragma unroll
        for (int r = 0; r < 8; ++r)
            rg[n][r] = sigmoidf_(ai[n][r] + bi + ah[n][r] + bh);
    }

    // n gate: tanh(i_n + r * h_n)
    gate_gemm(aggr, h, w_ih, w_hh, row0, m16, g, /*gg=*/2, ai, ah);
    v8f ng[4];
    #pragma unroll
    for (int n = 0; n < 4; ++n) {
        const int c = 2 * 64 + n * 16 + m16;
        const float bi = b_ih[c], bh = b_hh[c];
        #pragma unroll
        for (int r = 0; r < 8; ++r)
            ng[n][r] = tanhf((ai[n][r] + bi) + rg[n][r] * (ah[n][r] + bh));
    }

    // z gate + blend, store in place
    gate_gemm(aggr, h, w_ih, w_hh, row0, m16, g, /*gg=*/1, ai, ah);
    #pragma unroll
    for (int n = 0; n < 4; ++n) {
        const int c = 1 * 64 + n * 16 + m16;
        const float bi = b_ih[c], bh = b_hh[c];
        const int ncol = n * 16 + m16;
        #pragma unroll
        for (int r = 0; r < 8; ++r) {
            const float z = sigmoidf_(ai[n][r] + bi + ah[n][r] + bh);
            const int M = r + 8 * g;
            float* hp = h + (size_t)(row0 + M) * HID + ncol;
            const float hv = *hp;
            *hp = (1.0f - z) * ng[n][r] + z * hv;
        }
    }
}

// -------------------------------------------------------------------
// out[i] = sigmoid(h[idx[i]] . W_lin + b_lin)
// -------------------------------------------------------------------
__global__ void out_kernel(const float* __restrict__ h, const int* __restrict__ idx,
                           const float* __restrict__ Wlin, const float* __restrict__ blin,
                           float* __restrict__ out) {
    const int i = blockIdx.x * blockDim.x + threadIdx.x;
    if (i >= N_IDX) return;
    const int node = idx[i];
    const float4* hp = (const float4*)(h + (size_t)node * HID);
    const float4* wp = (const float4*)Wlin;
    float s = blin[0];
    #pragma unroll
    for (int j = 0; j < HID / 4; ++j) {
        const float4 hv = hp[j];
        const float4 wv = wp[j];
        s = fmaf(hv.x, wv.x, s);
        s = fmaf(hv.y, wv.y, s);
        s = fmaf(hv.z, wv.z, s);
        s = fmaf(hv.w, wv.w, s);
    }
    out[i] = sigmoidf_(s);
}

extern "C" void kernel_launch(void* const* d_in, const int* in_sizes, int n_in,
                              void* d_out, int out_size, void* d_ws, size_t ws_size,
                              hipStream_t stream) {
    const float* x     = (const float*)d_in[0];
    const int*   ei    = (const int*)d_in[1];
    const int*   idx   = (const int*)d_in[2];
    const float* W_red = (const float*)d_in[3];
    const float* b_red = (const float*)d_in[4];
    const float* W_g   = (const float*)d_in[5];
    const float* w_ih  = (const float*)d_in[6];
    const float* w_hh  = (const float*)d_in[7];
    const float* b_ih  = (const float*)d_in[8];
    const float* b_hh  = (const float*)d_in[9];
    const float* W_lin = (const float*)d_in[10];
    const float* b_lin = (const float*)d_in[11];
    float* out = (float*)d_out;

    float* h    = (float*)d_ws;                       // 100000*64 f32
    float* m    = h + (size_t)N_NODES * HID;          // 100000*64 f32
    float* aggr = m + (size_t)N_NODES * HID;          // 100000*64 f32

    const int* src = ei;
    const int* dst = ei + N_EDGES;

    const int gemm_blocks = (NTILES + WPB - 1) / WPB;               // 1563
    const int zero_n4     = N_NODES * HID / 4;                      // 1.6M
    const int zero_blocks = (zero_n4 + 255) / 256;
    const long long scat_threads = (long long)N_EDGES * 16;
    const int scat_blocks = (int)((scat_threads + 255) / 256);      // 50000
    const int out_blocks  = (N_IDX + 255) / 256;

    reduce_kernel<<<gemm_blocks, TPB, 0, stream>>>(x, W_red, b_red, h);

    for (int t = 0; t < N_STEPS; ++t) {
        mstep_kernel<<<gemm_blocks, TPB, 0, stream>>>(h, W_g + (size_t)t * HID * HID, m);
        zero_kernel<<<zero_blocks, 256, 0, stream>>>((float4*)aggr, zero_n4);
        scatter_kernel<<<scat_blocks, 256, 0, stream>>>(m, src, dst, aggr);
        gru_kernel<<<gemm_blocks, TPB, 0, stream>>>(aggr, h, w_ih, w_hh, b_ih, b_hh);
    }

    out_kernel<<<out_blocks, 256, 0, stream>>>(h, idx, W_lin, b_lin, out);
}